// Combine_STFT_63333587747232
// MI455X (gfx1250) — compile-verified
//
#include <hip/hip_runtime.h>

// Combine-STFT channel interleave:
//   X: (8, 16, 513, 1024) f32  ->  out: (8, 8, 513, 1024) complex64
//   real = X[:,0::2], imag = X[:,1::2], out = real + i*imag (interleaved f32 pairs)
//
// Pure bandwidth kernel: 269 MB in + 269 MB out, 0 FLOPs.
// Floor at 23.3 TB/s HBM ≈ 23 us. Strategy: fully dense vector VMEM with
// non-temporal cache hints (stream-through, working set > L2).
//
// Key identity: with plane = b*8 + c2 (0..63), P = 513*1024 floats:
//   real channel base = 2*plane*P,  imag base = (2*plane+1)*P,
//   output float base = 2*plane*P, element p -> output offset 2p.
// So each of the 64 output planes is a zip of two contiguous input streams.

typedef __attribute__((ext_vector_type(2))) float v2f;
typedef __attribute__((ext_vector_type(4))) float v4f;

#define PLANE_F (513 * 1024)  // floats per (h,w) plane == complex per output plane

__global__ __launch_bounds__(256) void
interleave_ri_kernel(const float* __restrict__ X, float* __restrict__ out) {
    const int plane = blockIdx.y;                       // 0..63  (b*8 + c2)
    const size_t realBase = (size_t)(2 * plane) * PLANE_F;
    const size_t imagBase = realBase + PLANE_F;
    const size_t outBase  = (size_t)plane * (2 * (size_t)PLANE_F);

    // Each block covers 1024 complex elements: 2 chunks of 512, 2 complex/lane.
    const int p0 = blockIdx.x * 1024 + threadIdx.x * 2;

#pragma unroll
    for (int half = 0; half < 2; ++half) {
        const int p = p0 + half * 512;

        // Dense b64 loads: lane L reads 8 bytes at byte-offset 8*L -> fully packed.
        const v2f* rp = (const v2f*)(X + realBase + p);
        const v2f* ip = (const v2f*)(X + imagBase + p);
        v2f r = __builtin_nontemporal_load(rp);
        v2f i = __builtin_nontemporal_load(ip);

        // Zip: {r0, i0, r1, i1}
        v4f o;
        o.x = r.x; o.y = i.x; o.z = r.y; o.w = i.y;

        // Dense b128 store: lane L writes 16 bytes at byte-offset 16*L -> fully packed.
        __builtin_nontemporal_store(o, (v4f*)(out + outBase + 2 * (size_t)p));
    }
}

extern "C" void kernel_launch(void* const* d_in, const int* in_sizes, int n_in,
                              void* d_out, int out_size, void* d_ws, size_t ws_size,
                              hipStream_t stream) {
    (void)in_sizes; (void)n_in; (void)out_size; (void)d_ws; (void)ws_size;
    const float* X = (const float*)d_in[0];
    float* out = (float*)d_out;

    // 64 planes (y), 513 blocks of 1024 complex each (x): 513*1024 = PLANE_F exactly.
    dim3 grid(513, 64, 1);
    dim3 block(256, 1, 1);
    interleave_ri_kernel<<<grid, block, 0, stream>>>(X, out);
}